// OmniAttention_47442208751828
// MI455X (gfx1250) — compile-verified
//
#include <hip/hip_runtime.h>
#include <math.h>

// ---- problem constants (from reference setup) ----
#define HW    3136   // 56*56
#define CIN   512
#define ACH   32     // attention channels
#define BATCH 64
#define COUT  512

typedef __attribute__((ext_vector_type(2))) float v2f;
typedef __attribute__((ext_vector_type(4))) float v4f;
typedef __attribute__((ext_vector_type(8))) float v8f;

// ---------------------------------------------------------------------------
// Kernel 1: global average pool over H*W.  Pure HBM-bandwidth kernel:
// one wave32 per (b,c) row of 3136 floats, b128 loads with NON-TEMPORAL
// hint (411 MB single-use stream > 192 MB L2 -> don't thrash L2), shuffle
// reduction.  Roofline floor ~17.6us @ 23.3 TB/s.
// ---------------------------------------------------------------------------
__global__ __launch_bounds__(256) void omni_pool(const float* __restrict__ x,
                                                 float* __restrict__ pooled) {
  const int wave = threadIdx.x >> 5;
  const int lane = threadIdx.x & 31;
  const int row  = blockIdx.x * 8 + wave;          // [0, 64*512)
  const v4f* __restrict__ xr = (const v4f*)(x + (size_t)row * HW);

  v4f acc = {0.f, 0.f, 0.f, 0.f};
  for (int i = lane; i < HW / 4; i += 32) {        // 784 float4 per row
    v4f v = __builtin_nontemporal_load(&xr[i]);    // TH=NT streaming load
    acc += v;
  }
  float s = (acc.x + acc.y) + (acc.z + acc.w);
#pragma unroll
  for (int off = 16; off >= 1; off >>= 1)
    s += __shfl_xor(s, off, 32);
  if (lane == 0) pooled[row] = s * (1.0f / (float)HW);
}

// ---------------------------------------------------------------------------
// Kernel 2: fused  h = ReLU(BN(pooled @ fc_w^T)); out = sigmoid(h @ W^T + b)
// for both channel and filter attention.  All GEMMs via
// V_WMMA_F32_16X16X4_F32 (fp32-precise matrix pipe), one workgroup, 8 waves.
// ---------------------------------------------------------------------------
__global__ __launch_bounds__(256) void omni_att(
    const float* __restrict__ pooled,                      // [64,512]
    const float* __restrict__ fc_w,                        // [32,512]
    const float* __restrict__ bn_gamma, const float* __restrict__ bn_beta,
    const float* __restrict__ bn_mean,  const float* __restrict__ bn_var,
    const float* __restrict__ ch_w, const float* __restrict__ ch_b,   // [512,32],[512]
    const float* __restrict__ f_w,  const float* __restrict__ f_b,    // [512,32],[512]
    float* __restrict__ out) {                             // [2,64,512] concat
  __shared__ float hbuf[BATCH * ACH];                      // 8 KB

  const int tid    = threadIdx.x;
  const int wave   = tid >> 5;
  const int lane   = tid & 31;
  const int lane16 = lane & 15;
  const int koff   = (lane >> 4) * 2;   // A/B frag K offset: lanes 16-31 hold K+2,K+3

  // ---------------- Stage A: h[64,32] = pooled @ fc_w^T, BN+ReLU ----------
  {
    const int mtile = wave & 3;         // 4 M-tiles of 16 rows
    const int ntile = wave >> 2;        // 2 N-tiles of 16 cols
    const int m = mtile * 16 + lane16;
    const int n = ntile * 16 + lane16;
    const float* __restrict__ arow = pooled + m * CIN;
    const float* __restrict__ brow = fc_w   + n * CIN;

    v8f c = {};
    for (int k = 0; k < CIN; k += 4) {
      v2f a, b;
      a.x = arow[k + koff]; a.y = arow[k + koff + 1];
      b.x = brow[k + koff]; b.y = brow[k + koff + 1];
      c = __builtin_amdgcn_wmma_f32_16x16x4_f32(false, a, false, b,
                                                (short)0, c, false, false);
    }
    // Fold BatchNorm (eval stats) + ReLU, column n
    const float scale = bn_gamma[n] * rsqrtf(bn_var[n] + 1e-5f);
    const float shift = bn_beta[n] - bn_mean[n] * scale;
    const int rbase = mtile * 16 + ((lane >> 4) << 3);  // C layout: M=r / r+8
#pragma unroll
    for (int r = 0; r < 8; ++r) {
      float v = c[r] * scale + shift;
      v = v > 0.f ? v : 0.f;
      hbuf[(rbase + r) * ACH + n] = v;
    }
  }
  __syncthreads();

  // ------- Stage B: out[o] = sigmoid(h @ W_o^T + b_o), o in {ch, filt} ----
  {
    const int o     = wave >> 2;        // 0: channel att, 1: filter att
    const int mtile = wave & 3;
    const float* __restrict__ W    = o ? f_w : ch_w;
    const float* __restrict__ bias = o ? f_b : ch_b;
    float* __restrict__ outp = out + o * (BATCH * COUT);

    const int m = mtile * 16 + lane16;
    v2f afr[8];                          // h A-fragments, K=32 in 8 steps of 4
#pragma unroll
    for (int ks = 0; ks < 8; ++ks) {
      const int k = ks * 4 + koff;
      afr[ks].x = hbuf[m * ACH + k];
      afr[ks].y = hbuf[m * ACH + k + 1];
    }
    const int rbase = mtile * 16 + ((lane >> 4) << 3);

    for (int ntile = 0; ntile < 32; ++ntile) {
      const int n = ntile * 16 + lane16;
      const float* __restrict__ brow = W + n * ACH;
      v8f c = {};
#pragma unroll
      for (int ks = 0; ks < 8; ++ks) {
        const int k = ks * 4 + koff;
        v2f b; b.x = brow[k]; b.y = brow[k + 1];
        c = __builtin_amdgcn_wmma_f32_16x16x4_f32(false, afr[ks], false, b,
                                                  (short)0, c, false, false);
      }
      const float bi = bias[n];
#pragma unroll
      for (int r = 0; r < 8; ++r) {
        float v = c[r] + bi;
        v = 1.0f / (1.0f + __expf(-v));
        outp[(rbase + r) * COUT + n] = v;
      }
    }
  }
}

extern "C" void kernel_launch(void* const* d_in, const int* in_sizes, int n_in,
                              void* d_out, int out_size, void* d_ws, size_t ws_size,
                              hipStream_t stream) {
  const float* x        = (const float*)d_in[0];
  const float* fc_w     = (const float*)d_in[1];
  const float* bn_gamma = (const float*)d_in[2];
  const float* bn_beta  = (const float*)d_in[3];
  const float* bn_mean  = (const float*)d_in[4];
  const float* bn_var   = (const float*)d_in[5];
  const float* ch_w     = (const float*)d_in[6];
  const float* ch_b     = (const float*)d_in[7];
  const float* f_w      = (const float*)d_in[8];
  const float* f_b      = (const float*)d_in[9];

  float* pooled = (float*)d_ws;          // 64*512 floats = 128 KB scratch

  // Stage 1: bandwidth-bound pooling (32768 rows, 1 wave/row, 8 waves/block)
  omni_pool<<<(BATCH * CIN) / 8, 256, 0, stream>>>(x, pooled);

  // Stage 2: fused WMMA attention head (tiny; single workgroup)
  omni_att<<<1, 256, 0, stream>>>(pooled, fc_w, bn_gamma, bn_beta, bn_mean,
                                  bn_var, ch_w, ch_b, f_w, f_b, (float*)d_out);
}